// SpatialHyperedge_59227599012104
// MI455X (gfx1250) — compile-verified
//
#include <hip/hip_runtime.h>
#include <hip/hip_bf16.h>
#include <math.h>

// ---------------------------------------------------------------------------
// Problem constants (from reference): B=8192, N=16, D=512
// out = [recon_loss (8192) | incidence_all (256)]
// ws  = [A matrix (256 floats) | extra scalar (1 float)]
// ---------------------------------------------------------------------------
#define BATCH   8192
#define NN      16
#define DD      512
#define GB      4        // batches per block: B-fragments reused G times
#define LDS_PITCH 516    // 512 + 4 pad -> conflict-free A-fragment reads

typedef __attribute__((ext_vector_type(2))) float v2f;
typedef __attribute__((ext_vector_type(8))) float v8f;

// ---------------------------------------------------------------------------
// Prep kernel: build A from incidence_m, write incidence_all = A + I,
// stash A and extra = sum_n(l1[n] + 0.001*l2[n]) into workspace.
// ---------------------------------------------------------------------------
__global__ __launch_bounds__(256)
void hyperedge_prep_kernel(const float* __restrict__ inc,
                           float* __restrict__ out_inc_all,
                           float* __restrict__ ws) {
    __shared__ float A[NN][NN];
    int t = threadIdx.x;

    A[t >> 4][t & 15] = 0.0f;
    __syncthreads();

    if (t < NN * (NN - 1)) {               // 240 entries
        int i = t / (NN - 1);
        int j = t % (NN - 1);
        float w = inc[t];
        w = (w > 0.0f) ? w : 0.0f;         // clamp(min=0) + strict >0 mask == relu
        int sj = j + ((j >= i) ? 1 : 0);   // skip diagonal
        A[i][sj] = w;
    }
    __syncthreads();

    if (t == 0) {
        float extra = 0.0f;
        for (int i = 0; i < NN; ++i) {
            float l1 = 0.0f, l2 = 0.0f;
            for (int j = 0; j < NN; ++j) { // diagonal is 0, harmless
                float w = A[i][j];
                l1 = fmaxf(l1, w);
                l2 += w * w;
            }
            extra += l1 + 0.001f * sqrtf(l2);
        }
        ws[NN * NN] = extra;
    }
    __syncthreads();

    float a = A[t >> 4][t & 15];
    ws[t] = a;
    out_inc_all[t] = a + (((t >> 4) == (t & 15)) ? 1.0f : 0.0f);
}

// ---------------------------------------------------------------------------
// Main kernel: one block (8 waves) per GB batches.
//   proj  = X_b @ r_proj      via v_wmma_f32_16x16x4_f32, K=512
//   diff  = proj - A @ X_b    (recon folded in with negated A-fragments)
//   loss  = 0.2 * sum_n ||diff_n||_2 + extra
// Each loaded r_proj B-fragment feeds GB WMMAs (batch-invariant reuse).
// ---------------------------------------------------------------------------
__global__ __launch_bounds__(256)
void hyperedge_batch_kernel(const float* __restrict__ X,
                            const float* __restrict__ R,
                            const float* __restrict__ ws,
                            float* __restrict__ out) {
    __shared__ float Xs[GB][NN][LDS_PITCH];   // 4 x 32KB (+pad) X tiles
    __shared__ float As[NN][NN];
    __shared__ float rowsum[GB][NN];

    const int b0 = blockIdx.x * GB;
    const int t  = threadIdx.x;

    // ---- stage X for GB batches (GB x 16x512 f32) into LDS, float4 loads ----
    {
        const float4* Xv = (const float4*)(X + (size_t)b0 * (NN * DD));
        #pragma unroll
        for (int it = 0; it < 8 * GB; ++it) {
            int f   = t + 256 * it;            // 0 .. GB*2048-1 float4's
            int g   = f >> 11;                 // 2048 float4 per batch
            int row = (f >> 7) & 15;           // 128 float4 per row
            int cv  = f & 127;
            float4 v = Xv[f];
            *(float4*)&Xs[g][row][cv << 2] = v;  // pad keeps 16B alignment
        }
    }
    As[t >> 4][t & 15] = ws[t];
    if (t < GB * NN) rowsum[t >> 4][t & 15] = 0.0f;
    __syncthreads();

    const int lane  = t & 31;
    const int wave  = t >> 5;
    const int m     = lane & 15;           // A-row / B-column index for this lane
    const int khalf = lane >> 4;           // 0: K pair {0,1}; 1: K pair {2,3}

    v8f sq[GB];                            // per-lane squared-diff accumulators
    #pragma unroll
    for (int g = 0; g < GB; ++g) sq[g] = (v8f){};

    #pragma unroll
    for (int tt = 0; tt < 4; ++tt) {
        const int j  = (wave << 2) + tt;   // tile 0..31
        const int c0 = j << 4;             // output column base
        const float* Bp = R + c0 + m;      // this lane's r_proj column

        v8f acc[GB];
        #pragma unroll
        for (int g = 0; g < GB; ++g) acc[g] = (v8f){};

        // ---- proj = X_b @ R over K = 512, step 4; B-frag shared by GB ----
        #pragma unroll 4
        for (int k = 0; k < DD; k += 4) {
            const int ka = k + (khalf << 1);
            v2f bb;
            bb.x = Bp[(size_t)ka << 9];                      // R[ka  ][c0+n]
            bb.y = Bp[((size_t)ka + 1) << 9];                // R[ka+1][c0+n]
            #pragma unroll
            for (int g = 0; g < GB; ++g) {
                v2f a = *(const v2f*)&Xs[g][m][ka];          // A frag: X_b row m
                acc[g] = __builtin_amdgcn_wmma_f32_16x16x4_f32(
                    false, a, false, bb, (short)0, acc[g], false, false);
            }
        }

        // ---- subtract recon: acc += (-A) @ X_b[:, c0:c0+16], K = 16 ----
        #pragma unroll
        for (int k = 0; k < NN; k += 4) {
            const int ka = k + (khalf << 1);
            v2f a;
            a.x = -As[m][ka];                                // VALU negation (f32
            a.y = -As[m][ka + 1];                            // WMMA has no A-neg)
            #pragma unroll
            for (int g = 0; g < GB; ++g) {
                v2f bb;
                bb.x = Xs[g][ka][c0 + m];
                bb.y = Xs[g][ka + 1][c0 + m];
                acc[g] = __builtin_amdgcn_wmma_f32_16x16x4_f32(
                    false, a, false, bb, (short)0, acc[g], false, false);
            }
        }

        #pragma unroll
        for (int g = 0; g < GB; ++g) {
            #pragma unroll
            for (int r = 0; r < 8; ++r) {
                float d = acc[g][r];
                sq[g][r] += d * d;
            }
        }
    }

    // ---- reduce over N (16 lanes of each wave32 half), merge waves in LDS ----
    #pragma unroll
    for (int g = 0; g < GB; ++g) {
        #pragma unroll
        for (int r = 0; r < 8; ++r) {
            float v = sq[g][r];
            v += __shfl_xor(v, 1);
            v += __shfl_xor(v, 2);
            v += __shfl_xor(v, 4);
            v += __shfl_xor(v, 8);         // stays within each 16-lane half
            if (m == 0) atomicAdd(&rowsum[g][r + (khalf << 3)], v);
        }
    }
    __syncthreads();

    if (t < GB) {
        float total = 0.0f;
        #pragma unroll
        for (int i = 0; i < NN; ++i) total += sqrtf(rowsum[t][i]);
        out[b0 + t] = 0.2f * total + ws[NN * NN];
    }
}

// ---------------------------------------------------------------------------
extern "C" void kernel_launch(void* const* d_in, const int* in_sizes, int n_in,
                              void* d_out, int out_size, void* d_ws, size_t ws_size,
                              hipStream_t stream) {
    (void)in_sizes; (void)n_in; (void)out_size; (void)ws_size;

    const float* X   = (const float*)d_in[0];   // [8192,16,512] f32
    const float* R   = (const float*)d_in[1];   // [512,512] f32
    const float* inc = (const float*)d_in[2];   // [16,15] f32
    float* out = (float*)d_out;                 // [8192 | 256]
    float* ws  = (float*)d_ws;                  // [256 A | 1 extra]

    hyperedge_prep_kernel<<<1, 256, 0, stream>>>(inc, out + BATCH, ws);
    hyperedge_batch_kernel<<<BATCH / GB, 256, 0, stream>>>(X, R, ws, out);
}